// NGramRepeatBlock_11931419149127
// MI455X (gfx1250) — compile-verified
//
#include <hip/hip_runtime.h>

// ---------------------------------------------------------------------------
// NGram repeat blocking for MI455X (gfx1250).
// Bandwidth-bound pass-through of lprobs (≈412 MB traffic → ~18 µs at 23.3
// TB/s) + a tiny window-match scatter. The copy uses the CDNA5 async
// global<->LDS data mover path (ASYNCcnt): builtins take non-const
// int4-typed pointers (AS1 global, AS3 LDS) + imm offset + imm cpol.
// ---------------------------------------------------------------------------

#if defined(__AMDGCN__) && \
    __has_builtin(__builtin_amdgcn_global_load_async_to_lds_b128) && \
    __has_builtin(__builtin_amdgcn_global_store_async_from_lds_b128) && \
    __has_builtin(__builtin_amdgcn_s_wait_asynccnt)
#define HAVE_GFX1250_ASYNC 1
#else
#define HAVE_GFX1250_ASYNC 0
#endif

// v4i32: the element type the async b128 builtins are declared with.
typedef int v4i __attribute__((vector_size(16)));
typedef __attribute__((address_space(1))) v4i g_v4i;   // global (AS1)
typedef __attribute__((address_space(3))) v4i l_v4i;   // LDS (AS3)

#define COPY_THREADS 256
#define COPY_STAGES  4

// ---------------------------------------------------------------------------
// Streaming copy: out[0..total) = in[0..total)  (floats), B128 granularity.
// ---------------------------------------------------------------------------
__global__ void __launch_bounds__(COPY_THREADS)
ngram_copy_kernel(const float* __restrict__ in_f,
                  float* __restrict__ out_f,
                  long long total)
{
    const long long n4     = total >> 2;                       // # of float4
    const long long tid    = (long long)blockIdx.x * blockDim.x + threadIdx.x;
    const long long stride = (long long)gridDim.x * blockDim.x;

    const float4* __restrict__ in4  = (const float4*)in_f;
    float4* __restrict__       out4 = (float4*)out_f;

    long long i = tid;

#if HAVE_GFX1250_ASYNC
    // Per-lane LDS staging: COPY_STAGES x 16B per thread (16 KB / workgroup).
    __shared__ alignas(16) float4 stage[COPY_STAGES][COPY_THREADS];
    const int t = threadIdx.x;

    // Keep COPY_STAGES B128 async loads in flight per lane (2 KB / wave),
    // then drain them with async stores straight out of LDS. ASYNCcnt
    // tracks both; wait-0 between phases gives load->store ordering on the
    // shared LDS slots.
    for (; i + (COPY_STAGES - 1) * stride < n4; i += COPY_STAGES * stride) {
#pragma unroll
        for (int s = 0; s < COPY_STAGES; ++s) {
            __builtin_amdgcn_global_load_async_to_lds_b128(
                (g_v4i*)(in4 + i + (long long)s * stride),
                (l_v4i*)&stage[s][t],
                0, 0);
        }
        __builtin_amdgcn_s_wait_asynccnt(0);
#pragma unroll
        for (int s = 0; s < COPY_STAGES; ++s) {
            __builtin_amdgcn_global_store_async_from_lds_b128(
                (g_v4i*)(out4 + i + (long long)s * stride),
                (l_v4i*)&stage[s][t],
                0, 0);
        }
        __builtin_amdgcn_s_wait_asynccnt(0);
    }
    // Remaining full float4 chunks (plain VGPR path).
    for (; i < n4; i += stride) {
        out4[i] = in4[i];
    }
#else
    // Fallback: plain unrolled B128 copy (4 loads in flight per lane).
    for (; i + 3 * stride < n4; i += 4 * stride) {
        float4 a = in4[i];
        float4 b = in4[i + stride];
        float4 c = in4[i + 2 * stride];
        float4 d = in4[i + 3 * stride];
        out4[i]              = a;
        out4[i + stride]     = b;
        out4[i + 2 * stride] = c;
        out4[i + 3 * stride] = d;
    }
    for (; i < n4; i += stride) {
        out4[i] = in4[i];
    }
#endif

    // Scalar tail (total % 4 floats).
    const long long tail_base = n4 << 2;
    const long long tail      = total - tail_base;
    if (tid < tail) {
        out_f[tail_base + tid] = in_f[tail_base + tid];
    }
}

// ---------------------------------------------------------------------------
// Mask kernel: for each row, ban tokens following earlier (n-1)-gram matches
// of the trailing (n-1)-gram. Scalars are read from device memory (they are
// 1-element device arrays; host cannot read them under graph capture).
// ---------------------------------------------------------------------------
__global__ void ngram_mask_kernel(const int* __restrict__ tokens,
                                  float* __restrict__ out,
                                  const int* __restrict__ bsz_p,
                                  const int* __restrict__ step_p,
                                  const int* __restrict__ beam_p,
                                  const int* __restrict__ n_p,
                                  long long out_elems)
{
    const int bsz  = bsz_p[0];
    const int step = step_p[0];
    const int beam = beam_p[0];
    const int n    = n_p[0];

    if (step + 2 - n <= 0) return;           // reference early-out

    const int       R = bsz * beam;
    const int       L = step + 1;            // tokens row stride
    const int       K = step - n + 2;        // number of window starts
    const long long V = out_elems / R;

    for (int row = blockIdx.x; row < R; row += gridDim.x) {
        const int* __restrict__ tok  = tokens + (long long)row * L;
        float* __restrict__     orow = out    + (long long)row * V;

        // trailing (n-1)-gram lives at tok[K .. K+n-2] (== step-n+2 .. step)
        for (int k = threadIdx.x; k < K; k += blockDim.x) {
            bool match = true;
            for (int j = 0; j < n - 1; ++j) {
                match &= (tok[k + j] == tok[K + j]);
            }
            if (match) {
                const int banned = tok[k + n - 1];
                if (banned >= 0 && (long long)banned < V) {
                    orow[banned] = -__builtin_inff();
                }
            }
        }
    }
}

// ---------------------------------------------------------------------------
// Launch. Input dict order: tokens, lprobs, bsz, step, beam_size,
// no_repeat_ngram_size. Output: (R, V) float32.
// ---------------------------------------------------------------------------
extern "C" void kernel_launch(void* const* d_in, const int* in_sizes, int n_in,
                              void* d_out, int out_size, void* d_ws, size_t ws_size,
                              hipStream_t stream)
{
    (void)in_sizes; (void)n_in; (void)d_ws; (void)ws_size;

    const int*   tokens = (const int*)d_in[0];
    const float* lprobs = (const float*)d_in[1];
    const int*   bsz_p  = (const int*)d_in[2];
    const int*   step_p = (const int*)d_in[3];
    const int*   beam_p = (const int*)d_in[4];
    const int*   n_p    = (const int*)d_in[5];
    float*       out    = (float*)d_out;

    const long long total = (long long)out_size;

    // Copy: enough blocks for full occupancy, grid-stride for the rest.
    long long n4 = total >> 2;
    long long want = (n4 + COPY_THREADS - 1) / COPY_THREADS;
    int blocks = (int)(want < 1 ? 1 : (want > 8192 ? 8192 : want));
    ngram_copy_kernel<<<blocks, COPY_THREADS, 0, stream>>>(lprobs, out, total);

    // Mask: grid-stride over rows (R known only on device).
    ngram_mask_kernel<<<1024, 256, 0, stream>>>(tokens, out,
                                                bsz_p, step_p, beam_p, n_p,
                                                total);
}